// GenAggSparse_8083128451335
// MI455X (gfx1250) — compile-verified
//
#include <hip/hip_runtime.h>

// ---------------------------------------------------------------------------
// GenAgg segmented power-mean aggregation for MI455X (gfx1250, wave32).
// E rows x 32 features scatter-reduced into S segments.
// D == 32 == wave32 lane count: lane d owns feature column d.
// ---------------------------------------------------------------------------

#define D 32
#define ROWS_PER_CHUNK 16              // 16 rows x 128B = 2KB per chunk
#define CHUNK_BYTES (ROWS_PER_CHUNK * D * 4)
static constexpr float kPLo = -1.98f;
static constexpr float kPHi =  1.98f;
static constexpr float kPiOver4 = 0.78539816339744830961f;

typedef float v4f __attribute__((ext_vector_type(4)));

// Workspace layout (bytes):
//   0   : u32 colmin_enc[32]
//   128 : u32 colmax_enc[32]
//   256 : f32 shift[32]
//   384 : f32 c[32]
//   512 : f32 p (1 value, padded)
//   640 : f32 Ncnt[S]
#define WS_CMIN  0
#define WS_CMAX  128
#define WS_SHIFT 256
#define WS_C     384
#define WS_P     512
#define WS_N     640

// Order-preserving float<->uint encoding so atomicMin/Max on unsigned works.
__device__ __forceinline__ unsigned enc_f32(float f) {
  unsigned u = __float_as_uint(f);
  return (u & 0x80000000u) ? ~u : (u | 0x80000000u);
}
__device__ __forceinline__ float dec_f32(unsigned u) {
  return __uint_as_float((u & 0x80000000u) ? (u & 0x7FFFFFFFu) : ~u);
}

// CDNA5 async global->LDS copy (ASYNCcnt-tracked). INST_OFFSET applies to BOTH
// the LDS destination and the global source address, so one immediate shifts
// the whole 512B transfer. NT hint: x is streamed once, keep it out of L2 so
// the atomic accumulator lines stay resident.
#define ASYNC_LDS_B128(ldsoff, voff, sbase, io)                               \
  asm volatile("global_load_async_to_lds_b128 %0, %1, %2 offset:" #io         \
               " th:TH_LOAD_NT"                                               \
               :: "v"(ldsoff), "v"(voff), "s"(sbase) : "memory")

#define WAIT_ASYNC(n) asm volatile("s_wait_asynccnt " #n ::: "memory")

// ---------------------------------------------------------------------------
// Pass 3 (defined first so the disasm snippet shows it): main accumulation.
// One wave owns a chunk of 16 contiguous rows (2KB), streamed into a private
// double-buffered LDS slot with four global_load_async_to_lds_b128 (512B each,
// lane L -> base + 16*L), overlapped with the compute/atomic phase of the
// previous chunk via s_wait_asynccnt. Per-wave buffers -> no barriers needed.
// ---------------------------------------------------------------------------
__device__ __forceinline__ void process_chunk(const float* __restrict__ buf,
                                              int c, int lane,
                                              const int* __restrict__ index,
                                              float p, float sh, float cv,
                                              float* __restrict__ dout,
                                              float* __restrict__ ncnt) {
  const int rowbase = c * ROWS_PER_CHUNK;
  int idxv = 0;
  if (lane < ROWS_PER_CHUNK) idxv = index[rowbase + lane];
#pragma unroll
  for (int i = 0; i < ROWS_PER_CHUNK; ++i) {
    int idx = __shfl(idxv, i, 32);
    float xs = buf[i * D + lane] - sh;
    float w  = expf(p * logf(xs) - cv);
    atomicAdd(&dout[idx * D + lane], w);
    if (lane == 0) atomicAdd(&ncnt[idx], 1.0f);
  }
}

__global__ __launch_bounds__(256) void k_accum(const float* __restrict__ x,
                                               const int* __restrict__ index,
                                               const float* __restrict__ shiftArr,
                                               const float* __restrict__ cArr,
                                               const float* __restrict__ pArr,
                                               float* __restrict__ dout,
                                               float* __restrict__ ncnt,
                                               int NC, int totalWaves) {
  __shared__ float tile[8][2][ROWS_PER_CHUNK * D];  // 8 waves x 2 x 2KB = 32KB
  const int lane = threadIdx.x & 31;
  const int wib  = threadIdx.x >> 5;
  const int gw   = (blockIdx.x * blockDim.x + threadIdx.x) >> 5;  // wave-uniform

  const float p  = pArr[0];
  const float sh = shiftArr[lane];
  const float cv = cArr[lane];

  // Low 32 bits of a generic LDS pointer are the wave-relative LDS byte offset.
  unsigned lds[2];
  lds[0] = (unsigned)(size_t)(const void*)&tile[wib][0][0] + (unsigned)(lane * 16);
  lds[1] = (unsigned)(size_t)(const void*)&tile[wib][1][0] + (unsigned)(lane * 16);

  int c = gw;
  if (c >= NC) return;  // uniform per wave

  {  // prime buffer 0 with chunk c: four 512B async transfers
    unsigned vo = (unsigned)c * (unsigned)CHUNK_BYTES + (unsigned)(lane * 16);
    ASYNC_LDS_B128(lds[0], vo, x, 0);
    ASYNC_LDS_B128(lds[0], vo, x, 512);
    ASYNC_LDS_B128(lds[0], vo, x, 1024);
    ASYNC_LDS_B128(lds[0], vo, x, 1536);
  }

  int cur = 0;
  int cn  = c + totalWaves;
  for (;;) {
    const bool more = (cn < NC);  // uniform per wave
    if (more) {
      unsigned vo = (unsigned)cn * (unsigned)CHUNK_BYTES + (unsigned)(lane * 16);
      ASYNC_LDS_B128(lds[cur ^ 1], vo, x, 0);
      ASYNC_LDS_B128(lds[cur ^ 1], vo, x, 512);
      ASYNC_LDS_B128(lds[cur ^ 1], vo, x, 1024);
      ASYNC_LDS_B128(lds[cur ^ 1], vo, x, 1536);
      WAIT_ASYNC(4);  // in-order: current buffer's 4 transfers are complete
    } else {
      WAIT_ASYNC(0);
    }
    process_chunk(&tile[wib][cur][0], c, lane, index, p, sh, cv, dout, ncnt);
    if (!more) break;
    c = cn;
    cn += totalWaves;
    cur ^= 1;
  }
}

// Remainder rows (E % ROWS_PER_CHUNK), if any: one wave, plain loads.
__global__ void k_accum_tail(const float* __restrict__ x,
                             const int* __restrict__ index,
                             const float* __restrict__ shiftArr,
                             const float* __restrict__ cArr,
                             const float* __restrict__ pArr,
                             float* __restrict__ dout,
                             float* __restrict__ ncnt,
                             int rowStart, int E) {
  const int lane = threadIdx.x;
  const float p = pArr[0], sh = shiftArr[lane], cv = cArr[lane];
  for (int r = rowStart; r < E; ++r) {
    int idx = index[r];
    float xs = x[r * D + lane] - sh;
    float w  = expf(p * logf(xs) - cv);
    atomicAdd(&dout[idx * D + lane], w);
    if (lane == 0) atomicAdd(&ncnt[idx], 1.0f);
  }
}

// ---------------------------------------------------------------------------
// Pass 0: initialize accumulators (d_out is poisoned by harness; we must zero
// it every call since it doubles as the segment-sum accumulator).
// ---------------------------------------------------------------------------
__global__ __launch_bounds__(256) void k_init(float* __restrict__ dout,
                                              float* __restrict__ ncnt,
                                              unsigned* __restrict__ cmin,
                                              unsigned* __restrict__ cmax,
                                              int SD, int S) {
  int i = blockIdx.x * blockDim.x + threadIdx.x;
  if (i < SD) dout[i] = 0.0f;
  if (i < S)  ncnt[i] = 0.0f;
  if (i < D)  { cmin[i] = 0xFFFFFFFFu; cmax[i] = 0u; }
}

// ---------------------------------------------------------------------------
// Pass 1: per-column min/max of x (non-temporal float4 sweep, LDS block
// reduce, 64 global atomics per block). Column of a float4 at flat idx i is
// ((4i) & 31)..+3 and stays fixed per thread because the grid stride is a
// multiple of 32 floats.
// ---------------------------------------------------------------------------
__global__ __launch_bounds__(256) void k_minmax(const v4f* __restrict__ x4,
                                                int n4,
                                                unsigned* __restrict__ gmin,
                                                unsigned* __restrict__ gmax) {
  __shared__ unsigned smin[D], smax[D];
  const int tid = threadIdx.x;
  if (tid < D) { smin[tid] = 0xFFFFFFFFu; smax[tid] = 0u; }
  __syncthreads();

  const int stride = gridDim.x * blockDim.x;
  float mn0 = INFINITY, mn1 = INFINITY, mn2 = INFINITY, mn3 = INFINITY;
  float mx0 = -INFINITY, mx1 = -INFINITY, mx2 = -INFINITY, mx3 = -INFINITY;
  for (int i = blockIdx.x * blockDim.x + tid; i < n4; i += stride) {
    v4f v = __builtin_nontemporal_load(&x4[i]);  // th:TH_LOAD_NT b128
    mn0 = fminf(mn0, v.x); mx0 = fmaxf(mx0, v.x);
    mn1 = fminf(mn1, v.y); mx1 = fmaxf(mx1, v.y);
    mn2 = fminf(mn2, v.z); mx2 = fmaxf(mx2, v.z);
    mn3 = fminf(mn3, v.w); mx3 = fmaxf(mx3, v.w);
  }
  const int b = (tid * 4) & (D - 1);
  atomicMin(&smin[b + 0], enc_f32(mn0)); atomicMax(&smax[b + 0], enc_f32(mx0));
  atomicMin(&smin[b + 1], enc_f32(mn1)); atomicMax(&smax[b + 1], enc_f32(mx1));
  atomicMin(&smin[b + 2], enc_f32(mn2)); atomicMax(&smax[b + 2], enc_f32(mx2));
  atomicMin(&smin[b + 3], enc_f32(mn3)); atomicMax(&smax[b + 3], enc_f32(mx3));
  __syncthreads();
  if (tid < D) {
    atomicMin(&gmin[tid], smin[tid]);
    atomicMax(&gmax[tid], smax[tid]);
  }
}

// ---------------------------------------------------------------------------
// Pass 2: scalar prep. p = tan(clip(p_param)*pi/4); shift = min - 1e-3;
// c = 0.5*(p*log(min_xs) + p*log(max_xs))  (midrange is symmetric, so the
// sign of p doesn't matter). min_xs/max_xs computed exactly as the reference
// rounds them: x_extreme - fl(x_min - 0.001).
// ---------------------------------------------------------------------------
__global__ void k_prep(const unsigned* __restrict__ cmin,
                       const unsigned* __restrict__ cmax,
                       const float* __restrict__ pp,
                       float* __restrict__ shiftArr,
                       float* __restrict__ cArr,
                       float* __restrict__ pOut) {
  int d = threadIdx.x;
  float p = tanf(fminf(fmaxf(pp[0], kPLo), kPHi) * kPiOver4);
  if (d == 0) pOut[0] = p;
  if (d < D) {
    float xmin = dec_f32(cmin[d]);
    float xmax = dec_f32(cmax[d]);
    float sh   = xmin - 0.001f;
    float mnxs = xmin - sh;   // ~0.001 with reference's rounding
    float mxxs = xmax - sh;
    shiftArr[d] = sh;
    cArr[d] = 0.5f * (p * logf(mnxs) + p * logf(mxxs));
  }
}

// ---------------------------------------------------------------------------
// Pass 4: finalize in place over [S, 32].
// lse = c + log(summed); Y = exp((1/p)*(lse - log N)); out = N^a * (Y + shift)
// Empty segments: summed==0 -> log -> -inf -> Y -> 0 (matches reference), N=1.
// ---------------------------------------------------------------------------
__global__ __launch_bounds__(256) void k_final(float* __restrict__ dout,
                                               const float* __restrict__ ncnt,
                                               const float* __restrict__ shiftArr,
                                               const float* __restrict__ cArr,
                                               const float* __restrict__ pArr,
                                               const float* __restrict__ aArr,
                                               int SD) {
  int i = blockIdx.x * blockDim.x + threadIdx.x;
  if (i >= SD) return;
  const int s = i >> 5;
  const int d = i & (D - 1);
  const float p = pArr[0];
  const float a = aArr[0];
  float n = ncnt[s];
  if (n == 0.0f) n = 1.0f;
  const float logn   = logf(n);
  const float summed = dout[i];
  const float lse    = cArr[d] + logf(summed);
  const float y      = expf((1.0f / p) * (lse - logn));
  const float na     = expf(a * logn);  // n >= 1
  dout[i] = na * (y + shiftArr[d]);
}

// ---------------------------------------------------------------------------
extern "C" void kernel_launch(void* const* d_in, const int* in_sizes, int n_in,
                              void* d_out, int out_size, void* d_ws, size_t ws_size,
                              hipStream_t stream) {
  const float* x     = (const float*)d_in[0];
  const float* pp    = (const float*)d_in[1];
  const float* aa    = (const float*)d_in[2];
  const int*   index = (const int*)d_in[3];
  // d_in[4] (dim_size) is implied by out_size.

  const int E  = in_sizes[3];
  const int SD = out_size;
  const int S  = SD / D;
  float* dout = (float*)d_out;

  char* ws = (char*)d_ws;
  unsigned* cmin  = (unsigned*)(ws + WS_CMIN);
  unsigned* cmax  = (unsigned*)(ws + WS_CMAX);
  float* shiftArr = (float*)(ws + WS_SHIFT);
  float* cArr     = (float*)(ws + WS_C);
  float* pArr     = (float*)(ws + WS_P);
  float* ncnt     = (float*)(ws + WS_N);

  // Pass 0: init accumulators.
  k_init<<<(SD + 255) / 256, 256, 0, stream>>>(dout, ncnt, cmin, cmax, SD, S);

  // Pass 1: column min/max.
  const int n4 = (E * D) / 4;
  k_minmax<<<1024, 256, 0, stream>>>((const v4f*)x, n4, cmin, cmax);

  // Pass 2: scalars.
  k_prep<<<1, 32, 0, stream>>>(cmin, cmax, pp, shiftArr, cArr, pArr);

  // Pass 3: segmented accumulation (async-LDS double-buffered streaming).
  const int NC = E / ROWS_PER_CHUNK;   // 2KB chunks of 16 rows
  const int blocks = 1024;             // 8192 waves
  const int totalWaves = blocks * 256 / 32;
  if (NC > 0) {
    k_accum<<<blocks, 256, 0, stream>>>(x, index, shiftArr, cArr, pArr,
                                        dout, ncnt, NC, totalWaves);
  }
  const int rem = E - NC * ROWS_PER_CHUNK;
  if (rem > 0) {
    k_accum_tail<<<1, 32, 0, stream>>>(x, index, shiftArr, cArr, pArr,
                                       dout, ncnt, NC * ROWS_PER_CHUNK, E);
  }

  // Pass 4: finalize in place.
  k_final<<<(SD + 255) / 256, 256, 0, stream>>>(dout, ncnt, shiftArr, cArr,
                                                pArr, aa, SD);
}